// TopKRouter_48576080118263
// MI455X (gfx1250) — compile-verified
//
#include <hip/hip_runtime.h>

typedef __attribute__((ext_vector_type(2))) float v2f;
typedef __attribute__((ext_vector_type(8))) float v8f;

#define N_TOK 8192
#define DIM   4096
#define NEXP  64
#define TOPK  2
#define CAP   160                         // ceil(1.25 * 8192 / 64)
#define NEC   (8192ull * 64ull * 160ull)  // N*E*C elements per big output

// ---------------------------------------------------------------------------
// Zero-fill the two big (N,E,C) outputs with 128-bit stores.
// ---------------------------------------------------------------------------
__global__ void __launch_bounds__(256) zero_kernel(float4* __restrict__ p, size_t n4) {
    size_t i      = (size_t)blockIdx.x * blockDim.x + threadIdx.x;
    size_t stride = (size_t)gridDim.x * blockDim.x;
    float4 z = make_float4(0.f, 0.f, 0.f, 0.f);
    for (; i < n4; i += stride) p[i] = z;
}

// ---------------------------------------------------------------------------
// logits = x @ gate_w^T  via V_WMMA_F32_16X16X4_F32.
// One wave computes a 16(row) x 64(expert) strip: 4 accumulators, K-step 4.
// A layout (16x4 f32): lanes 0-15 -> M=lane, K={0,1}; lanes 16-31 -> K={2,3}.
// B layout (4x16 f32): mirrored (lane&15 = N, half-wave selects K pair).
// C/D layout: VGPR r holds M=r (lanes 0-15) / M=r+8 (lanes 16-31), N=lane&15.
// ---------------------------------------------------------------------------
__global__ void __launch_bounds__(128) gemm_logits_kernel(const float* __restrict__ x,
                                                          const float* __restrict__ w,
                                                          float* __restrict__ logits) {
    const int lane = threadIdx.x & 31;
    const int wave = blockIdx.x * 4 + (threadIdx.x >> 5);   // 0..511 row tiles
    const int row0 = wave * 16;
    const int m    = lane & 15;
    const int kk   = (lane >> 4) * 2;

    const float* xrow = x + (size_t)(row0 + m) * DIM + kk;
    const float* w0   = w + (size_t)(m)      * DIM + kk;    // experts 0..15
    const float* w1   = w + (size_t)(m + 16) * DIM + kk;    // experts 16..31
    const float* w2   = w + (size_t)(m + 32) * DIM + kk;    // experts 32..47
    const float* w3   = w + (size_t)(m + 48) * DIM + kk;    // experts 48..63

    v8f acc0 = {}, acc1 = {}, acc2 = {}, acc3 = {};

    for (int k = 0; k < DIM; k += 4) {
        v2f a  = *(const v2f*)(xrow + k);
        v2f b0 = *(const v2f*)(w0 + k);
        v2f b1 = *(const v2f*)(w1 + k);
        v2f b2 = *(const v2f*)(w2 + k);
        v2f b3 = *(const v2f*)(w3 + k);
        acc0 = __builtin_amdgcn_wmma_f32_16x16x4_f32(false, a, false, b0, (short)0, acc0, false, false);
        acc1 = __builtin_amdgcn_wmma_f32_16x16x4_f32(false, a, false, b1, (short)0, acc1, false, false);
        acc2 = __builtin_amdgcn_wmma_f32_16x16x4_f32(false, a, false, b2, (short)0, acc2, false, false);
        acc3 = __builtin_amdgcn_wmma_f32_16x16x4_f32(false, a, false, b3, (short)0, acc3, false, false);
    }

    const int n     = lane & 15;
    const int rbase = (lane >> 4) * 8;
    for (int r = 0; r < 8; ++r) {
        size_t rowOff = (size_t)(row0 + rbase + r) * NEXP;
        logits[rowOff +      n] = acc0[r];
        logits[rowOff + 16 + n] = acc1[r];
        logits[rowOff + 32 + n] = acc2[r];
        logits[rowOff + 48 + n] = acc3[r];
    }
}

// ---------------------------------------------------------------------------
// Per-token softmax over 64 experts + top-2 (wave32 per token, 2 experts/lane).
// ---------------------------------------------------------------------------
__global__ void __launch_bounds__(256) softmax_topk_kernel(const float* __restrict__ logits,
                                                           float* __restrict__ probs_out,
                                                           float* __restrict__ vals,
                                                           int*   __restrict__ idx) {
    const int lane = threadIdx.x & 31;
    const int t    = blockIdx.x * 8 + (threadIdx.x >> 5);

    const float l0 = logits[(size_t)t * NEXP + lane];
    const float l1 = logits[(size_t)t * NEXP + lane + 32];

    float mx = fmaxf(l0, l1);
    for (int off = 16; off; off >>= 1) mx = fmaxf(mx, __shfl_xor(mx, off, 32));

    float e0 = __expf(l0 - mx), e1 = __expf(l1 - mx);
    float s = e0 + e1;
    for (int off = 16; off; off >>= 1) s += __shfl_xor(s, off, 32);

    float p0 = e0 / s, p1 = e1 / s;
    probs_out[(size_t)t * NEXP + lane]      = p0;
    probs_out[(size_t)t * NEXP + lane + 32] = p1;

    // top-1 (lowest index wins ties, matching jax.lax.top_k)
    float bv; int be;
    if (p0 >= p1) { bv = p0; be = lane; } else { bv = p1; be = lane + 32; }
    for (int off = 16; off; off >>= 1) {
        float ov = __shfl_xor(bv, off, 32);
        int   oe = __shfl_xor(be, off, 32);
        if (ov > bv || (ov == bv && oe < be)) { bv = ov; be = oe; }
    }
    // top-2: exclude top-1 expert (probs >= 0, so -1 is a safe sentinel)
    float q0 = (lane      == be) ? -1.0f : p0;
    float q1 = (lane + 32 == be) ? -1.0f : p1;
    float cv; int ce;
    if (q0 >= q1) { cv = q0; ce = lane; } else { cv = q1; ce = lane + 32; }
    for (int off = 16; off; off >>= 1) {
        float ov = __shfl_xor(cv, off, 32);
        int   oe = __shfl_xor(ce, off, 32);
        if (ov > cv || (ov == cv && oe < ce)) { cv = ov; ce = oe; }
    }

    if (lane == 0) {
        float vs = bv + cv;
        vals[t * TOPK + 0] = bv / vs;
        vals[t * TOPK + 1] = cv / vs;
        idx[t * TOPK + 0]  = be;
        idx[t * TOPK + 1]  = ce;
    }
}

// ---------------------------------------------------------------------------
// p[e] = mean over tokens of probs[:, e]   (one block per expert)
// ---------------------------------------------------------------------------
__global__ void __launch_bounds__(256) colmean_kernel(const float* __restrict__ probs,
                                                      float* __restrict__ p_mean) {
    __shared__ float red[256];
    const int e = blockIdx.x;
    float s = 0.f;
    for (int t = threadIdx.x; t < N_TOK; t += 256) s += probs[(size_t)t * NEXP + e];
    red[threadIdx.x] = s;
    __syncthreads();
    for (int off = 128; off; off >>= 1) {
        if (threadIdx.x < off) red[threadIdx.x] += red[threadIdx.x + off];
        __syncthreads();
    }
    if (threadIdx.x == 0) p_mean[e] = red[0] / (float)N_TOK;
}

// ---------------------------------------------------------------------------
// Sequential exclusive prefix count per expert over the flat (N*K) stream.
// Thread e scans all assignments in order (matches reference cumsum exactly).
// ---------------------------------------------------------------------------
__global__ void slot_assign_kernel(const int* __restrict__ idx,
                                   int* __restrict__ slot,
                                   int* __restrict__ fcount) {
    const int e = threadIdx.x;
    int cnt = 0;
    for (int i = 0; i < N_TOK * TOPK; ++i) {
        if (idx[i] == e) { slot[i] = cnt; cnt++; }
    }
    fcount[e] = cnt;   // uncapped total -> f[e] numerator
}

// ---------------------------------------------------------------------------
// Scatter kept assignments into dispatch_mask / combine_weights.
// ---------------------------------------------------------------------------
__global__ void __launch_bounds__(256) scatter_kernel(const float* __restrict__ vals,
                                                      const int* __restrict__ idx,
                                                      const int* __restrict__ slot,
                                                      float* __restrict__ mask_out,
                                                      float* __restrict__ comb_out) {
    const int i = blockIdx.x * blockDim.x + threadIdx.x;
    if (i >= N_TOK * TOPK) return;
    const int t = i >> 1;           // TOPK == 2
    const int e = idx[i];
    const int s = slot[i];
    if (s < CAP) {                  // keep iff prior count < capacity
        size_t base = ((size_t)t * NEXP + e) * CAP + s;
        mask_out[base] = 1.0f;
        comb_out[base] = vals[i];
    }
}

// ---------------------------------------------------------------------------
// load_loss = AUX_W * E * sum_e f[e] * p[e]
// ---------------------------------------------------------------------------
__global__ void loss_kernel(const int* __restrict__ fcount,
                            const float* __restrict__ p_mean,
                            float* __restrict__ loss_out) {
    if (threadIdx.x == 0 && blockIdx.x == 0) {
        float acc = 0.f;
        for (int e = 0; e < NEXP; ++e)
            acc += ((float)fcount[e] / (float)(N_TOK * TOPK)) * p_mean[e];
        *loss_out = 0.01f * (float)NEXP * acc;
    }
}

extern "C" void kernel_launch(void* const* d_in, const int* in_sizes, int n_in,
                              void* d_out, int out_size, void* d_ws, size_t ws_size,
                              hipStream_t stream) {
    const float* x = (const float*)d_in[0];   // (8192, 4096) f32
    const float* w = (const float*)d_in[1];   // (64, 4096)   f32

    float* out       = (float*)d_out;
    float* mask_out  = out;                                   // (N,E,C)
    float* comb_out  = out + NEC;                             // (N,E,C)
    float* probs_out = out + 2 * NEC;                         // (N,E)
    float* loss_out  = out + 2 * NEC + (size_t)N_TOK * NEXP;  // scalar

    // Workspace layout
    float* logits = (float*)d_ws;                             // N*E
    float* vals   = logits + (size_t)N_TOK * NEXP;            // N*K
    int*   idx    = (int*)(vals + N_TOK * TOPK);              // N*K
    int*   slot   = idx + N_TOK * TOPK;                       // N*K
    int*   fcount = slot + N_TOK * TOPK;                      // E
    float* p_mean = (float*)(fcount + NEXP);                  // E

    // 1) zero the 671 MB of dispatch/combine output (dominant memory traffic)
    zero_kernel<<<4096, 256, 0, stream>>>((float4*)out, (2 * NEC) / 4);

    // 2) logits GEMM on the WMMA path (512 waves, 4 waves/block)
    gemm_logits_kernel<<<128, 128, 0, stream>>>(x, w, logits);

    // 3) softmax + top-2 per token (wave per token)
    softmax_topk_kernel<<<N_TOK / 8, 256, 0, stream>>>(logits, probs_out, vals, idx);

    // 4) mean prob per expert (aux loss input)
    colmean_kernel<<<NEXP, 256, 0, stream>>>(probs_out, p_mean);

    // 5) sequential capacity/slot assignment (one thread per expert)
    slot_assign_kernel<<<1, NEXP, 0, stream>>>(idx, slot, fcount);

    // 6) scatter kept assignments
    scatter_kernel<<<(N_TOK * TOPK + 255) / 256, 256, 0, stream>>>(vals, idx, slot,
                                                                   mask_out, comb_out);

    // 7) scalar aux loss
    loss_kernel<<<1, 1, 0, stream>>>(fcount, p_mean, loss_out);
}